// GNNLayer_60155311947936
// MI455X (gfx1250) — compile-verified
//
#include <hip/hip_runtime.h>
#include <math.h>

#define N_NODES  20000
#define N_EDGES  320000
#define E_TOT    (N_EDGES + N_NODES)
#define HEADS    4
#define CH       64
#define HC       256   // HEADS*CH
#define NEG_SLOPE 0.2f
#define BN_EPS    1e-5f

typedef __attribute__((ext_vector_type(2))) float v2f;
typedef __attribute__((ext_vector_type(8))) float v8f;

// ---------------------------------------------------------------- utilities
__global__ void fill_kernel(float* __restrict__ p, int n, float v) {
    int i = blockIdx.x * blockDim.x + threadIdx.x;
    if (i < n) p[i] = v;
}

__device__ __forceinline__ void atomicMaxF(float* addr, float v) {
    // standard monotone-bit-pattern trick: works across mixed signs
    if (v >= 0.0f) atomicMax((int*)addr, __float_as_int(v));
    else           atomicMin((unsigned int*)addr, __float_as_uint(v));
}

// ---------------------------------------------------------------- 1) xl = x @ W
// x:[20000,64]  W:[64,256]  xl:[20000,256]
// 512 threads = 16 wave32; block owns 16 rows of x (staged in LDS),
// wave w owns output columns [w*16, w*16+16). K loop: 16 steps of 4 via
// V_WMMA_F32_16X16X4_F32 (fp32-exact).
__global__ __launch_bounds__(512) void gemm_wmma_kernel(
        const float* __restrict__ x, const float* __restrict__ W,
        float* __restrict__ xl) {
    __shared__ float sx[16][65];   // stride 65 -> conflict-free column reads
    const int m0 = blockIdx.x * 16;
    const int t  = threadIdx.x;
    for (int i = t; i < 16 * 64; i += 512)
        sx[i >> 6][i & 63] = x[(size_t)(m0 + (i >> 6)) * 64 + (i & 63)];
    __syncthreads();

    const int wave  = t >> 5;
    const int lane  = t & 31;
    const int n0    = wave * 16;
    const int mlane = lane & 15;
    const int khalf = (lane >> 4) << 1;   // lanes 0-15 -> K{0,1}, 16-31 -> K{2,3}

    v8f acc = {};
    for (int k0 = 0; k0 < 64; k0 += 4) {
        const int kb = k0 + khalf;
        v2f a, b;
        // A 16x4 f32 layout (ISA 7.12.2): lane = M, vgpr0/1 = K pair
        a.x = sx[mlane][kb];
        a.y = sx[mlane][kb + 1];
        // B 4x16 f32 layout (transpose symmetric): lane = N, vgpr0/1 = K pair
        b.x = W[(size_t)kb       * HC + n0 + mlane];
        b.y = W[(size_t)(kb + 1) * HC + n0 + mlane];
        acc = __builtin_amdgcn_wmma_f32_16x16x4_f32(
                  false, a, false, b, (short)0, acc, false, false);
    }
    // C/D layout: vgpr r -> M = r (lanes 0-15) / r+8 (lanes 16-31), N = lane&15
    const int mrow = m0 + ((lane >> 4) << 3);
#pragma unroll
    for (int r = 0; r < 8; ++r)
        xl[(size_t)(mrow + r) * HC + n0 + mlane] = acc[r];
}

// ---------------------------------------------------------------- 2) logits
// one thread per (node, head); fused logit_i and logit_j (shares xl/emb loads)
__global__ void logits_kernel(
        const float* __restrict__ xl,  const float* __restrict__ emb,
        const float* __restrict__ ati, const float* __restrict__ atj,
        const float* __restrict__ aei, const float* __restrict__ aej,
        float* __restrict__ logit_i,   float* __restrict__ logit_j) {
    int idx = blockIdx.x * blockDim.x + threadIdx.x;
    if (idx >= N_NODES * HEADS) return;
    const int n = idx >> 2, h = idx & 3;
    const float4* xv = (const float4*)(xl  + (size_t)n * HC + h * CH);
    const float4* ev = (const float4*)(emb + (size_t)n * CH);
    const float4* ai = (const float4*)(ati + h * CH);
    const float4* aj = (const float4*)(atj + h * CH);
    const float4* ei = (const float4*)(aei + h * CH);
    const float4* ej = (const float4*)(aej + h * CH);
    float si = 0.f, sj = 0.f;
#pragma unroll 4
    for (int k = 0; k < 16; ++k) {
        float4 xq = xv[k], eq = ev[k];
        float4 a = ai[k], b = aj[k], c = ei[k], d = ej[k];
        si += xq.x*a.x + xq.y*a.y + xq.z*a.z + xq.w*a.w
            + eq.x*c.x + eq.y*c.y + eq.z*c.z + eq.w*c.w;
        sj += xq.x*b.x + xq.y*b.y + xq.z*b.z + xq.w*b.w
            + eq.x*d.x + eq.y*d.y + eq.z*d.z + eq.w*d.w;
    }
    logit_i[idx] = si;
    logit_j[idx] = sj;
}

// ---------------------------------------------------------------- 3) alpha raw + segment max
__global__ void alpha_kernel(
        const int* __restrict__ ei,
        const float* __restrict__ logit_i, const float* __restrict__ logit_j,
        float* __restrict__ alphaBuf, float* __restrict__ amax) {
    int e = blockIdx.x * blockDim.x + threadIdx.x;
    if (e >= E_TOT) return;
    int s, d;
    if (e < N_EDGES) { s = ei[e]; d = ei[N_EDGES + e]; }
    else             { s = d = e - N_EDGES; }   // self loops appended
    const float4 li = *(const float4*)(logit_i + (size_t)d * 4);
    const float4 lj = *(const float4*)(logit_j + (size_t)s * 4);
    float a[4] = { li.x + lj.x, li.y + lj.y, li.z + lj.z, li.w + lj.w };
#pragma unroll
    for (int h = 0; h < 4; ++h) {
        float v = a[h];
        v = (v >= 0.f) ? v : NEG_SLOPE * v;            // leaky relu
        alphaBuf[(size_t)e * 4 + h] = v;
        atomicMaxF(&amax[(size_t)d * 4 + h], v);
    }
}

// ---------------------------------------------------------------- 4) exp + denom
__global__ void expdenom_kernel(
        const int* __restrict__ ei, float* __restrict__ alphaBuf,
        const float* __restrict__ amax, float* __restrict__ denom) {
    int e = blockIdx.x * blockDim.x + threadIdx.x;
    if (e >= E_TOT) return;
    int d = (e < N_EDGES) ? ei[N_EDGES + e] : (e - N_EDGES);
#pragma unroll
    for (int h = 0; h < 4; ++h) {
        float v = expf(alphaBuf[(size_t)e * 4 + h] - amax[(size_t)d * 4 + h]);
        alphaBuf[(size_t)e * 4 + h] = v;
        atomicAdd(&denom[(size_t)d * 4 + h], v);
    }
}

// ---------------------------------------------------------------- 5) weighted scatter-add
// block-per-edge, 256 threads = (head,channel). Fully coalesced gather/scatter.
__global__ __launch_bounds__(256) void scatter_kernel(
        const int* __restrict__ ei, const float* __restrict__ alphaBuf,
        const float* __restrict__ denom, const float* __restrict__ xl,
        float* __restrict__ out3) {
    const int e = blockIdx.x;
    const int t = threadIdx.x;
    int s, d;
    if (e < N_EDGES) { s = ei[e]; d = ei[N_EDGES + e]; }
    else             { s = d = e - N_EDGES; }
    const int h = t >> 6;
    const float w = alphaBuf[(size_t)e * 4 + h] /
                    (denom[(size_t)d * 4 + h] + 1e-16f);
    const float m = w * xl[(size_t)s * HC + t];
    atomicAdd(&out3[(size_t)d * HC + t], m);
}

// ---------------------------------------------------------------- 6) head mean + bias + BN partials
__global__ __launch_bounds__(256) void meanbn_kernel(
        const float* __restrict__ out3, const float* __restrict__ bias,
        float* __restrict__ out_pre, float* __restrict__ gsum,
        float* __restrict__ gsumsq) {
    __shared__ float ssum[256];
    __shared__ float ssq[256];
    const int t   = threadIdx.x;
    const int idx = blockIdx.x * 256 + t;       // over N*CH; c = idx&63
    const int n = idx >> 6, c = idx & 63;
    const float* row = out3 + (size_t)n * HC;
    float pre = 0.25f * (row[c] + row[64 + c] + row[128 + c] + row[192 + c])
              + bias[c];
    out_pre[idx] = pre;
    ssum[t] = pre;
    ssq[t]  = pre * pre;
    __syncthreads();
    if (t < 64) {   // 4 nodes per block share channel c == t
        float s = ssum[t] + ssum[t + 64] + ssum[t + 128] + ssum[t + 192];
        float q = ssq[t]  + ssq[t + 64]  + ssq[t + 128]  + ssq[t + 192];
        atomicAdd(&gsum[t],  s);
        atomicAdd(&gsumsq[t], q);
    }
}

// ---------------------------------------------------------------- 7) BN normalize + ReLU
__global__ void finalize_kernel(
        const float* __restrict__ out_pre, const float* __restrict__ gsum,
        const float* __restrict__ gsumsq, const float* __restrict__ gamma,
        const float* __restrict__ beta, float* __restrict__ out) {
    int idx = blockIdx.x * blockDim.x + threadIdx.x;
    if (idx >= N_NODES * CH) return;
    const int c = idx & 63;
    const float inv_n = 1.0f / (float)N_NODES;
    const float mu  = gsum[c] * inv_n;
    const float var = gsumsq[c] * inv_n - mu * mu;
    float y = (out_pre[idx] - mu) * rsqrtf(var + BN_EPS) * gamma[c] + beta[c];
    out[idx] = fmaxf(y, 0.0f);
}

// ---------------------------------------------------------------- launch
extern "C" void kernel_launch(void* const* d_in, const int* in_sizes, int n_in,
                              void* d_out, int out_size, void* d_ws, size_t ws_size,
                              hipStream_t stream) {
    const float* x      = (const float*)d_in[0];
    const int*   eidx   = (const int*)  d_in[1];
    const float* emb    = (const float*)d_in[2];
    const float* W      = (const float*)d_in[3];
    const float* att_i  = (const float*)d_in[4];
    const float* att_j  = (const float*)d_in[5];
    const float* aem_i  = (const float*)d_in[6];
    const float* aem_j  = (const float*)d_in[7];
    const float* bias   = (const float*)d_in[8];
    const float* gamma  = (const float*)d_in[9];
    const float* beta   = (const float*)d_in[10];
    float* out = (float*)d_out;

    // workspace layout (floats)
    float* ws = (float*)d_ws;
    size_t o = 0;
    float* xl       = ws + o; o += (size_t)N_NODES * HC;   // 5,120,000
    float* logit_i  = ws + o; o += (size_t)N_NODES * HEADS;
    float* logit_j  = ws + o; o += (size_t)N_NODES * HEADS;
    float* amax     = ws + o; o += (size_t)N_NODES * HEADS;
    float* denom    = ws + o; o += (size_t)N_NODES * HEADS;
    float* alphaBuf = ws + o; o += (size_t)E_TOT   * HEADS;
    float* out3     = ws + o; o += (size_t)N_NODES * HC;   // 5,120,000
    float* out_pre  = ws + o; o += (size_t)N_NODES * CH;
    float* gsum     = ws + o; o += 64;
    float* gsumsq   = ws + o; o += 64;

    // re-init accumulators every call (atomics)
    fill_kernel<<<(N_NODES * HC + 255) / 256, 256, 0, stream>>>(out3, N_NODES * HC, 0.0f);
    fill_kernel<<<(N_NODES * HEADS + 255) / 256, 256, 0, stream>>>(denom, N_NODES * HEADS, 0.0f);
    fill_kernel<<<(N_NODES * HEADS + 255) / 256, 256, 0, stream>>>(amax, N_NODES * HEADS, -INFINITY);
    fill_kernel<<<1, 128, 0, stream>>>(gsum, 128, 0.0f);   // gsum+gsumsq contiguous

    gemm_wmma_kernel<<<N_NODES / 16, 512, 0, stream>>>(x, W, xl);

    logits_kernel<<<(N_NODES * HEADS + 255) / 256, 256, 0, stream>>>(
        xl, emb, att_i, att_j, aem_i, aem_j, logit_i, logit_j);

    alpha_kernel<<<(E_TOT + 255) / 256, 256, 0, stream>>>(
        eidx, logit_i, logit_j, alphaBuf, amax);

    expdenom_kernel<<<(E_TOT + 255) / 256, 256, 0, stream>>>(
        eidx, alphaBuf, amax, denom);

    scatter_kernel<<<E_TOT, 256, 0, stream>>>(eidx, alphaBuf, denom, xl, out3);

    meanbn_kernel<<<(N_NODES * CH) / 256, 256, 0, stream>>>(
        out3, bias, out_pre, gsum, gsumsq);

    finalize_kernel<<<(N_NODES * CH + 255) / 256, 256, 0, stream>>>(
        out_pre, gsum, gsumsq, gamma, beta, out);
}